// Efficient8BitALU_Shift_7945689497931
// MI455X (gfx1250) — compile-verified
//
#include <hip/hip_runtime.h>
#include <cstdint>

typedef float v4f __attribute__((ext_vector_type(4)));

#define WAVES_PER_BLOCK 8
#define TOKENS_PER_WAVE 8

__global__ __launch_bounds__(256) void alu_shift_stream_kernel(
    const float* __restrict__ x, float* __restrict__ out, int numTokens)
{
    // 8 waves * 2 buffers * 64 float4 (1 KB) = 16 KB LDS
    __shared__ v4f smem[WAVES_PER_BLOCK * 2 * 64];

    const int lane = threadIdx.x & 31;
    const int wid  = threadIdx.x >> 5;
    const int gw   = blockIdx.x * WAVES_PER_BLOCK + wid;
    const int t0   = gw * TOKENS_PER_WAVE;

    // Low 32 bits of the generic pointer to a shared variable are the LDS byte
    // offset (flat-LDS aperture keeps the DS offset in addr[31:0]).
    const uint32_t ldsWaveBase =
        (uint32_t)(uintptr_t)(&smem[wid * 2 * 64]);
    const uint32_t laneByte = (uint32_t)lane * 16u;
    const uint64_t srcBase  = (uint64_t)(uintptr_t)x;

    // Issue one token's 1 KB as two async b128 DMA loads into LDS.
    // The immediate offset is added to both the LDS address and the global address.
    // Input is read exactly once and exceeds L2 -> non-temporal load hint.
    auto issue_async = [&](int buf, int tok) {
        uint32_t ldsA = ldsWaveBase + (uint32_t)buf * 1024u + laneByte;
        uint32_t voff = (uint32_t)tok * 1024u + laneByte;  // < 4 GB for this problem
        asm volatile(
            "global_load_async_to_lds_b128 %0, %1, %2 th:TH_LOAD_NT\n\t"
            "global_load_async_to_lds_b128 %0, %1, %2 offset:512 th:TH_LOAD_NT"
            :
            : "v"(ldsA), "v"(voff), "s"(srcBase)
            : "memory");
    };

    if (t0 < numTokens) issue_async(0, t0);

    for (int i = 0; i < TOKENS_PER_WAVE; ++i) {
        const int t = t0 + i;
        if (t >= numTokens) break;

        // WAR guard: slot (i+1)&1 was read (ds_load) in iteration i-1; make sure
        // those LDS reads completed before the async engine overwrites the slot.
        asm volatile("s_wait_dscnt 0x0" ::: "memory");

        const bool haveNext = (i + 1 < TOKENS_PER_WAVE) && (t + 1 < numTokens);
        if (haveNext) {
            issue_async((i + 1) & 1, t + 1);
            // 4 async loads may be outstanding; <=2 guarantees token i's 2 are done
            asm volatile("s_wait_asynccnt 0x2" ::: "memory");
        } else {
            asm volatile("s_wait_asynccnt 0x0" ::: "memory");
        }

        // Lane l owns features [8l, 8l+8)
        const int base = wid * 2 * 64 + (i & 1) * 64 + lane * 2;
        v4f a = smem[base];
        v4f b = smem[base + 1];

        // Pack per-lane predicate byte for features 8l..8l+7
        const int gbase = lane * 8;
        unsigned pb = 0u;
#pragma unroll
        for (int j = 0; j < 4; ++j) {
            const int gi = gbase + j;
            const bool p = (gi == 0) ? (a[j] >= 0.5f) : (a[j] > 0.5f); // MARK uses >=
            pb |= (unsigned)p << j;
        }
#pragma unroll
        for (int j = 0; j < 4; ++j) {
            pb |= (unsigned)(b[j] > 0.5f) << (4 + j);
        }

        // Features 0..63 live in lanes 0..7 (8 predicate bits each).
        // 3-step OR-butterfly within groups of 8 lanes gives lane 0 the full
        // masks; readfirstlane broadcasts AND scalarizes the uniform decode.
        unsigned lo32 = (lane < 4) ? (pb << (8 * lane)) : 0u;
        unsigned hi32 = (lane >= 4 && lane < 8) ? (pb << (8 * (lane - 4))) : 0u;
#pragma unroll
        for (int m = 4; m >= 1; m >>= 1) {
            lo32 |= __shfl_xor(lo32, m, 32);
            hi32 |= __shfl_xor(hi32, m, 32);
        }
        const unsigned slo = (unsigned)__builtin_amdgcn_readfirstlane((int)lo32);
        const unsigned shi = (unsigned)__builtin_amdgcn_readfirstlane((int)hi32);

        // ---- wave-uniform decode (SALU) ----
        const bool mark = (slo & 1u) != 0u;
        const bool shl  = (slo & 2u) != 0u;
        const bool shr  = (!shl) && ((slo & 4u) != 0u);    // elif ordering

        const unsigned mlo = (slo >> 16) & 0xFFFFu;        // ALU_LO  (16..31)
        const unsigned mhi = shi & 0xFFFFu;                // ALU_HI  (32..47)
        const unsigned msh = (shi >> 16) & 0xFFFFu;        // SHIFT   (48..63)
        const int lo = mlo ? __builtin_ctz(mlo) : 0;       // first-hot, 0 if none
        const int hi = mhi ? __builtin_ctz(mhi) : 0;
        const int sh = msh ? __builtin_ctz(msh) : 0;

        const int  value  = hi * 16 + lo;
        const bool active = mark && (shl || shr);
        const int  res    = shl ? ((value << sh) & 255) : (value >> sh);
        const float addv  = active ? 2.0f : 0.0f;
        const int f0 = 96  + (res & 15);    // OUTPUT_LO slot
        const int f1 = 112 + (res >> 4);    // OUTPUT_HI slot

#pragma unroll
        for (int j = 0; j < 4; ++j) {
            const int gi = gbase + j;
            if (gi == f0 || gi == f1) a[j] += addv;
        }
#pragma unroll
        for (int j = 0; j < 4; ++j) {
            const int gi = gbase + 4 + j;
            if (gi == f0 || gi == f1) b[j] += addv;
        }

        v4f* outp = (v4f*)(out + (size_t)t * 256) + lane * 2;
        __builtin_nontemporal_store(a, &outp[0]);
        __builtin_nontemporal_store(b, &outp[1]);
    }
}

extern "C" void kernel_launch(void* const* d_in, const int* in_sizes, int n_in,
                              void* d_out, int out_size, void* d_ws, size_t ws_size,
                              hipStream_t stream) {
    (void)n_in; (void)d_ws; (void)ws_size; (void)out_size;
    const float* x = (const float*)d_in[0];
    float* out = (float*)d_out;

    const int numTokens = in_sizes[0] / 256;                          // B*S = 262144
    const int wavesNeeded = (numTokens + TOKENS_PER_WAVE - 1) / TOKENS_PER_WAVE;
    const int blocks = (wavesNeeded + WAVES_PER_BLOCK - 1) / WAVES_PER_BLOCK;

    alu_shift_stream_kernel<<<dim3(blocks), dim3(WAVES_PER_BLOCK * 32), 0, stream>>>(
        x, out, numTokens);
}